// EdgeConvE_8564164788721
// MI455X (gfx1250) — compile-verified
//
#include <hip/hip_runtime.h>
#include <cstdint>

typedef __attribute__((ext_vector_type(2))) float v2f;
typedef __attribute__((ext_vector_type(8))) float v8f;

#define V_NODES 1024
#define C_FEAT  64
#define E_FEAT  16
#define K_NBR   32
#define OUT_F   128
#define KDIM    144                       // 2C + E
#define NODES_PER_WG 4
#define M_ROWS  (NODES_PER_WG * K_NBR)    // 128 edge rows per workgroup
#define M_TILES (M_ROWS / 16)             // 8
#define FEAT_STRIDE 148                   // padded LDS row stride (floats): bank-conflict-free, 16B aligned

__global__ __launch_bounds__(256)
void edgeconv_wmma_kernel(const float* __restrict__ xf,   // [B*V, C]
                          const float* __restrict__ ea,   // [B, V, V, E]
                          const float* __restrict__ W,    // [144, 128]
                          const float* __restrict__ bias, // [128]
                          const int*   __restrict__ src,  // [Ne]
                          const int*   __restrict__ dst,  // [Ne]
                          float*       __restrict__ out)  // [B*V, 128]
{
    __shared__ float feat[M_ROWS * FEAT_STRIDE];          // 75,776 bytes (CDNA5: 320KB/WGP)

    const int tid   = threadIdx.x;
    const int node0 = blockIdx.x * NODES_PER_WG;
    const int edge0 = node0 * K_NBR;

    // ---- Stage feature tile [128 x 144] = [x_v | x_vp - x_v | e] into LDS ----
    {
        const int row  = tid >> 1;        // 0..127, 2 threads per edge row
        const int part = tid & 1;
        const int g    = edge0 + row;
        const int s    = src[g];
        const int d    = dst[g];
        const float* xv = xf + (size_t)s * C_FEAT;
        const float* xu = xf + (size_t)d * C_FEAT;
        float* fr = &feat[row * FEAT_STRIDE];
        #pragma unroll
        for (int c = part * 32; c < part * 32 + 32; c += 4) {
            float4 a  = *(const float4*)(xv + c);
            float4 bq = *(const float4*)(xu + c);
            *(float4*)(fr + c) = a;
            float4 dl = make_float4(bq.x - a.x, bq.y - a.y, bq.z - a.z, bq.w - a.w);
            *(float4*)(fr + C_FEAT + c) = dl;
        }
        // e = edge_attributes[s/V, s%V, d%V] -> flat index (s*V + d%V)*E
        const int ui = d & (V_NODES - 1);
        const float* ep = ea + ((size_t)s * V_NODES + ui) * E_FEAT;
        #pragma unroll
        for (int c = part * 8; c < part * 8 + 8; c += 4) {
            float4 ev = *(const float4*)(ep + c);
            *(float4*)(fr + 2 * C_FEAT + c) = ev;
        }
    }
    __syncthreads();

    // ---- GEMM via V_WMMA_F32_16X16X4_F32: each wave = one 16-column slice ----
    const int lane = tid & 31;
    const int wave = tid >> 5;            // 0..7
    const int col0 = wave * 16;
    const int hcol = lane & 15;           // A: M index / B: N index
    const int krow = (lane < 16) ? 0 : 2; // upper half-wave takes K+2, K+3

    v8f acc[M_TILES] = {};                // 8 M-tiles (rows 0..127)
    #pragma unroll 4
    for (int k0 = 0; k0 < KDIM; k0 += 4) {
        // B operand: W[k0+krow .. k0+krow+1][col0+hcol] (L1/L2 resident, 73 KB total)
        const float* wp = W + (size_t)(k0 + krow) * OUT_F + col0 + hcol;
        v2f bop; bop.x = wp[0]; bop.y = wp[OUT_F];
        #pragma unroll
        for (int mt = 0; mt < M_TILES; ++mt) {
            const float* ap = &feat[(mt * 16 + hcol) * FEAT_STRIDE + k0 + krow];
            v2f aop; aop.x = ap[0]; aop.y = ap[1];
            acc[mt] = __builtin_amdgcn_wmma_f32_16x16x4_f32(
                false, aop, false, bop, (short)0, acc[mt], false, false);
        }
    }

    // ---- segment-max over the 32 edge rows of each node, then +bias, ReLU ----
    // D layout: VGPR g -> row mt*16+g (lanes 0-15) / mt*16+8+g (lanes 16-31),
    //           column = col0 + (lane & 15).
    float m[NODES_PER_WG];
    #pragma unroll
    for (int n = 0; n < NODES_PER_WG; ++n) {
        float mv = -__builtin_huge_valf();
        #pragma unroll
        for (int g = 0; g < 8; ++g)
            mv = fmaxf(mv, fmaxf(acc[2 * n][g], acc[2 * n + 1][g]));
        mv = fmaxf(mv, __shfl_xor(mv, 16, 32));   // merge the two half-wave row groups
        m[n] = mv;
    }

    if (lane < 16) {
        const float bb = bias[col0 + hcol];
        #pragma unroll
        for (int n = 0; n < NODES_PER_WG; ++n)
            out[(size_t)(node0 + n) * OUT_F + col0 + hcol] = fmaxf(m[n] + bb, 0.0f);
    }
}

extern "C" void kernel_launch(void* const* d_in, const int* in_sizes, int n_in,
                              void* d_out, int out_size, void* d_ws, size_t ws_size,
                              hipStream_t stream) {
    const float* xf  = (const float*)d_in[0];  // node_features [4,1024,64]
    const float* ea  = (const float*)d_in[1];  // edge_attributes [4,1024,1024,16]
    const float* W   = (const float*)d_in[2];  // [144,128]
    const float* b   = (const float*)d_in[3];  // [128]
    const int*   src = (const int*)d_in[4];
    const int*   dst = (const int*)d_in[5];
    float* out = (float*)d_out;                // [4,1024,128]

    const int numNodes = 4 * 1024;
    dim3 grid(numNodes / NODES_PER_WG);        // 1024 workgroups
    edgeconv_wmma_kernel<<<grid, 256, 0, stream>>>(xf, ea, W, b, src, dst, out);
}